// GAT_67765993996807
// MI455X (gfx1250) — compile-verified
//
#include <hip/hip_runtime.h>

// Problem constants (match reference)
namespace {
constexpr int B_ = 32, S_ = 100, U_ = 256, NG_ = 5, G_ = 20, D_ = 10, H_ = 5, K_ = 2;
constexpr float SCALE = 0.70710678118654752440f;  // 1/sqrt(K), K=2
constexpr float NEG_SLOPE = 0.3f;
}

typedef __attribute__((ext_vector_type(2))) float v2f;
typedef __attribute__((ext_vector_type(8))) float v8f;

// Async global->LDS copy, 4 bytes per lane (GLOBAL_LOAD_ASYNC_TO_LDS_B32).
// ldsoff = wave-relative LDS byte offset (low 32 bits of generic shared ptr);
// gptr   = per-lane 64-bit global address. Tracked by ASYNCcnt.
__device__ __forceinline__ void async_copy_b32(unsigned ldsoff, const void* gptr) {
    asm volatile("global_load_async_to_lds_b32 %0, %1, off"
                 :: "v"(ldsoff), "v"(gptr) : "memory");
}
__device__ __forceinline__ void wait_asynccnt0() {
    asm volatile("s_wait_asynccnt 0x0" ::: "memory");
}

// One workgroup per (b,g). 256 threads = 8 wave32 waves.
// Each wave handles 10 (head, 16-row query tile) units: 8*10 = 80 = H*(U/16).
__global__ __launch_bounds__(256, 1)
void GAT_segattn_wmma(const float* __restrict__ x,
                      const float* __restrict__ Wq, const float* __restrict__ bq,
                      const float* __restrict__ Wk, const float* __restrict__ bk,
                      const float* __restrict__ Wv, const float* __restrict__ bv,
                      const float* __restrict__ Wo, const float* __restrict__ bo,
                      float* __restrict__ out)
{
    __shared__ float h_lds[NG_][U_];              // x segment, transposed view
    __shared__ float qkv_lds[3][H_][U_][2];       // Q,K,V for all heads
    __shared__ float ctx_lds[H_][U_][2];          // attention context
    __shared__ float weff[3][NG_][H_][K_];        // W[2i]+W[2i+1] (repeat-cols collapse)
    __shared__ float bqkv[3][H_][K_];
    __shared__ float wo_l[H_][K_][D_];
    __shared__ float bo_l[D_];
    __shared__ float red[256];

    const int bg = blockIdx.x;
    const int b  = bg / G_;
    const int g  = bg % G_;
    const int t  = threadIdx.x;

    // ---- phase 0: async-DMA x rows into LDS; stage collapsed weights ----
    const float* xrow = x + ((size_t)b * S_ + (size_t)g * NG_) * U_;
#pragma unroll
    for (int i = 0; i < NG_; ++i)
        async_copy_b32((unsigned)(uintptr_t)&h_lds[i][t], xrow + i * U_ + t);

    if (t < NG_ * H_ * K_) {                      // 50 threads stage Wq/Wk/Wv
        const int i = t / (H_ * K_);
        const int h = (t / K_) % H_;
        const int k = t % K_;
        const int i0 = ((g * D_ + 2 * i    ) * H_ + h) * K_ + k;
        const int i1 = ((g * D_ + 2 * i + 1) * H_ + h) * K_ + k;
        weff[0][i][h][k] = Wq[i0] + Wq[i1];
        weff[1][i][h][k] = Wk[i0] + Wk[i1];
        weff[2][i][h][k] = Wv[i0] + Wv[i1];
        if (i == 0) {
            const int bi = (g * H_ + h) * K_ + k;
            bqkv[0][h][k] = bq[bi];
            bqkv[1][h][k] = bk[bi];
            bqkv[2][h][k] = bv[bi];
        }
    }
    if (t < H_ * K_ * D_) {                       // 100 threads stage Wo
        const int h = t / (K_ * D_);
        const int k = (t / D_) % K_;
        const int d = t % D_;
        wo_l[h][k][d] = Wo[((g * H_ + h) * K_ + k) * D_ + d];
    }
    if (t < D_) bo_l[t] = bo[g * D_ + t];
    wait_asynccnt0();                             // async tiles resident in LDS
    __syncthreads();

    // ---- phase 1: Q/K/V projections; thread t == token u ----
    float hv[NG_];
#pragma unroll
    for (int i = 0; i < NG_; ++i) hv[i] = h_lds[i][t];
#pragma unroll
    for (int p = 0; p < 3; ++p)
#pragma unroll
        for (int h = 0; h < H_; ++h) {
            float a0 = bqkv[p][h][0], a1 = bqkv[p][h][1];
#pragma unroll
            for (int i = 0; i < NG_; ++i) {
                a0 = fmaf(hv[i], weff[p][i][h][0], a0);
                a1 = fmaf(hv[i], weff[p][i][h][1], a1);
            }
            v2f qa; qa.x = a0; qa.y = a1;
            *(v2f*)&qkv_lds[p][h][t][0] = qa;
        }
    __syncthreads();

    // ---- phase 2: attention via V_WMMA_F32_16X16X4_F32 + online softmax ----
    const int wave = t >> 5;
    const int lane = t & 31;
    const int lm   = lane & 15;
    const bool lo  = lane < 16;                   // lanes 16..31 carry K=2,3 (zero pad)
    const float SL2E = SCALE * 1.44269504088896340736f;  // scale * log2(e)

#pragma unroll 1
    for (int j = 0; j < 10; ++j) {
        const int unit = wave * 10 + j;           // 0..79
        const int h  = unit >> 4;                 // head
        const int qt = unit & 15;                 // query tile

        // B = Q^T tile (4x16): lane n holds q[qt*16+n][0..1]; hi half = zeros
        v2f Bq;
        {
            const v2f qv = *(const v2f*)&qkv_lds[0][h][qt * 16 + lm][0];
            Bq.x = lo ? qv.x : 0.0f;
            Bq.y = lo ? qv.y : 0.0f;
        }

        float m = -3.0e38f, l = 0.0f, a0 = 0.0f, a1 = 0.0f;

        for (int vt = 0; vt < 16; ++vt) {
            // A = K tile (16x4): lane m holds k[vt*16+m][0..1]; hi half = zeros
            v2f Ak;
            {
                const v2f kv = *(const v2f*)&qkv_lds[1][h][vt * 16 + lm][0];
                Ak.x = lo ? kv.x : 0.0f;
                Ak.y = lo ? kv.y : 0.0f;
            }
            v8f C = {};
            // C[m][n] = dot(k[vt*16+m], q[qt*16+n])  (scores transposed)
            C = __builtin_amdgcn_wmma_f32_16x16x4_f32(false, Ak, false, Bq,
                                                      (short)0, C, false, false);

            // lane owns query column qt*16+lm; its 8 C regs are key rows
            // vt*16 + r (+8 for hi half). Online softmax (base-2) update:
            float s[8], tmax = m;
#pragma unroll
            for (int r = 0; r < 8; ++r) { s[r] = C[r] * SL2E; tmax = fmaxf(tmax, s[r]); }
            const float corr = __builtin_amdgcn_exp2f(m - tmax);
            l *= corr; a0 *= corr; a1 *= corr;
            const int rbase = vt * 16 + (lo ? 0 : 8);
#pragma unroll
            for (int r = 0; r < 8; ++r) {
                const float p  = __builtin_amdgcn_exp2f(s[r] - tmax);
                const v2f  vv = *(const v2f*)&qkv_lds[2][h][rbase + r][0];
                l += p;
                a0 = fmaf(p, vv.x, a0);
                a1 = fmaf(p, vv.y, a1);
            }
            m = tmax;
        }

        // merge the two half-wave partial softmax states (row halves differ by +8)
        const float m_o  = __shfl_xor(m,  16, 32);
        const float l_o  = __shfl_xor(l,  16, 32);
        const float a0_o = __shfl_xor(a0, 16, 32);
        const float a1_o = __shfl_xor(a1, 16, 32);
        const float mn = fmaxf(m, m_o);
        const float c0 = __builtin_amdgcn_exp2f(m - mn);
        const float c1 = __builtin_amdgcn_exp2f(m_o - mn);
        const float inv = 1.0f / (l * c0 + l_o * c1);
        const float o0 = (a0 * c0 + a0_o * c1) * inv;
        const float o1 = (a1 * c0 + a1_o * c1) * inv;
        if (lo) {
            ctx_lds[h][qt * 16 + lm][0] = o0;
            ctx_lds[h][qt * 16 + lm][1] = o1;
        }
    }
    __syncthreads();

    // ---- phase 3: output projection, LeakyReLU, mean over (d,u) ----
    float c[H_][K_];
#pragma unroll
    for (int h = 0; h < H_; ++h) {
        const v2f cv = *(const v2f*)&ctx_lds[h][t][0];
        c[h][0] = cv.x; c[h][1] = cv.y;
    }
    float sum_u = 0.0f;
#pragma unroll
    for (int d = 0; d < D_; ++d) {
        float o = bo_l[d];
#pragma unroll
        for (int h = 0; h < H_; ++h) {
            o = fmaf(c[h][0], wo_l[h][0][d], o);
            o = fmaf(c[h][1], wo_l[h][1][d], o);
        }
        o = (o >= 0.0f) ? o : NEG_SLOPE * o;
        sum_u += o;
    }
    red[t] = sum_u;
    __syncthreads();
#pragma unroll
    for (int off = 128; off > 0; off >>= 1) {
        if (t < off) red[t] += red[t + off];
        __syncthreads();
    }
    if (t == 0) out[bg] = red[0] * (1.0f / (float)(U_ * D_));
}

extern "C" void kernel_launch(void* const* d_in, const int* in_sizes, int n_in,
                              void* d_out, int out_size, void* d_ws, size_t ws_size,
                              hipStream_t stream) {
    (void)in_sizes; (void)n_in; (void)out_size; (void)d_ws; (void)ws_size;
    const float* x  = (const float*)d_in[0];
    const float* Wq = (const float*)d_in[1];
    const float* bq = (const float*)d_in[2];
    const float* Wk = (const float*)d_in[3];
    const float* bk = (const float*)d_in[4];
    const float* Wv = (const float*)d_in[5];
    const float* bv = (const float*)d_in[6];
    const float* Wo = (const float*)d_in[7];
    const float* bo = (const float*)d_in[8];
    float* out = (float*)d_out;

    GAT_segattn_wmma<<<dim3(B_ * G_), dim3(256), 0, stream>>>(
        x, Wq, bq, Wk, bk, Wv, bv, Wo, bo, out);
}